// BiConv_12094627906069
// MI455X (gfx1250) — compile-verified
//
#include <hip/hip_runtime.h>

typedef __attribute__((ext_vector_type(2))) float v2f;
typedef __attribute__((ext_vector_type(8))) float v8f;

#define CHANNELS 64

// ---------------------------------------------------------------------------
// Kernel 1: agg_out = agg_back = x   (float4 copy)
// ---------------------------------------------------------------------------
__global__ void biconv_init(const float* __restrict__ x,
                            float* __restrict__ agg_out,
                            float* __restrict__ agg_back,
                            long long n4) {
    long long gid = (long long)blockIdx.x * blockDim.x + threadIdx.x;
    if (gid >= n4) return;
    float4 v = ((const float4*)x)[gid];
    ((float4*)agg_out)[gid]  = v;
    ((float4*)agg_back)[gid] = v;
}

// ---------------------------------------------------------------------------
// Kernel 2: scatter-add both directions.
// thread -> (edge, 4-channel group). float4 gather + 4x atomicAdd f32 each way.
// ---------------------------------------------------------------------------
__global__ void biconv_scatter(const float* __restrict__ x,
                               const int* __restrict__ src,
                               const int* __restrict__ tgt,
                               float* __restrict__ agg_out,
                               float* __restrict__ agg_back,
                               long long total /* = E*16 */) {
    long long gid = (long long)blockIdx.x * blockDim.x + threadIdx.x;
    if (gid >= total) return;
    long long e  = gid >> 4;
    int       c4 = ((int)gid & 15) << 2;
    long long s = (long long)src[e];
    long long t = (long long)tgt[e];

    float4 xs = *(const float4*)(x + s * CHANNELS + c4);
    float4 xt = *(const float4*)(x + t * CHANNELS + c4);

    float* po = agg_out + t * CHANNELS + c4;   // x[src] accumulated at tgt
    atomicAdd(po + 0, xs.x);
    atomicAdd(po + 1, xs.y);
    atomicAdd(po + 2, xs.z);
    atomicAdd(po + 3, xs.w);

    float* pb = agg_back + s * CHANNELS + c4;  // x[tgt] accumulated at src
    atomicAdd(pb + 0, xt.x);
    atomicAdd(pb + 1, xt.y);
    atomicAdd(pb + 2, xt.z);
    atomicAdd(pb + 3, xt.w);
}

// ---------------------------------------------------------------------------
// Kernel 3: out = (norm*agg_out) @ w_out + (norm_t*agg_back) @ w_back
// One wave32 per 16x16 output tile, V_WMMA_F32_16X16X4_F32, K accumulated
// over 64 (w_out) + 64 (w_back) = 32 WMMAs per tile.
//
// A 16x4 f32 layout: lanes 0-15 -> M=0..15 (kb=0), lanes 16-31 -> M=0..15
// (kb=2); VGPR0=K+kb, VGPR1=K+kb+1.  B 4x16 mirrored: lane n = lane&15,
// VGPR0=row K+kb, VGPR1=row K+kb+1.  D: col n = lane&15, row = j + 8*(lane>>4).
// ---------------------------------------------------------------------------
__global__ __launch_bounds__(256)
void biconv_wmma_gemm(const float* __restrict__ agg_out,
                      const float* __restrict__ agg_back,
                      const float* __restrict__ norm,
                      const float* __restrict__ norm_t,
                      const float* __restrict__ w_out,
                      const float* __restrict__ w_back,
                      float* __restrict__ out,
                      int N, int numTiles /* = ceil(N/16)*4 */) {
    const int wave = threadIdx.x >> 5;
    const int lane = threadIdx.x & 31;
    const int tile = blockIdx.x * 8 + wave;
    if (tile >= numTiles) return;            // wave-uniform: EXEC stays full

    const int rowTile = tile >> 2;
    const int colTile = tile & 3;

    const int m    = lane & 15;              // A row within tile
    const int half = lane >> 4;              // 0 or 1
    const int kb   = half << 1;              // K sub-offset {0,2}
    const int n    = lane & 15;              // B/D column within tile
    const int col  = colTile * 16 + n;

    int row  = rowTile * 16 + m;
    int rowc = row < N ? row : (N - 1);      // clamp loads, keep EXEC full

    const float sc_o = norm[rowc];
    const float sc_b = norm_t[rowc];
    const float* __restrict__ arow_o = agg_out  + (long long)rowc * CHANNELS;
    const float* __restrict__ arow_b = agg_back + (long long)rowc * CHANNELS;

    v8f acc = {};

    // --- K loop over w_out contribution ---
#pragma unroll
    for (int k = 0; k < CHANNELS; k += 4) {
        float2 av = *(const float2*)(arow_o + k + kb);
        v2f a; a.x = av.x * sc_o; a.y = av.y * sc_o;
        v2f b;
        b.x = w_out[(k + kb)     * CHANNELS + col];
        b.y = w_out[(k + kb + 1) * CHANNELS + col];
        acc = __builtin_amdgcn_wmma_f32_16x16x4_f32(
            false, a, false, b, (short)0, acc, false, false);
    }
    // --- K loop over w_back contribution ---
#pragma unroll
    for (int k = 0; k < CHANNELS; k += 4) {
        float2 av = *(const float2*)(arow_b + k + kb);
        v2f a; a.x = av.x * sc_b; a.y = av.y * sc_b;
        v2f b;
        b.x = w_back[(k + kb)     * CHANNELS + col];
        b.y = w_back[(k + kb + 1) * CHANNELS + col];
        acc = __builtin_amdgcn_wmma_f32_16x16x4_f32(
            false, a, false, b, (short)0, acc, false, false);
    }

    // --- store D: VGPR j -> row (j + 8*half) of tile, col `col` ---
#pragma unroll
    for (int j = 0; j < 8; ++j) {
        int mrow = rowTile * 16 + j + half * 8;
        if (mrow < N)
            out[(long long)mrow * CHANNELS + col] = acc[j];
    }
}

// ---------------------------------------------------------------------------
extern "C" void kernel_launch(void* const* d_in, const int* in_sizes, int n_in,
                              void* d_out, int out_size, void* d_ws, size_t ws_size,
                              hipStream_t stream) {
    const float* x       = (const float*)d_in[0];
    const int*   sources = (const int*)d_in[1];
    const int*   targets = (const int*)d_in[2];
    const float* norm    = (const float*)d_in[3];
    const float* norm_t  = (const float*)d_in[4];
    const float* w_out   = (const float*)d_in[5];
    const float* w_back  = (const float*)d_in[6];
    float*       out     = (float*)d_out;

    const int N = in_sizes[3];             // norm has N elements
    const long long E = in_sizes[1];       // edge count

    float* agg_out  = (float*)d_ws;
    float* agg_back = agg_out + (long long)N * CHANNELS;

    // 1) init agg buffers = x
    {
        long long n4 = (long long)N * (CHANNELS / 4);
        int threads = 256;
        long long blocks = (n4 + threads - 1) / threads;
        biconv_init<<<(unsigned)blocks, threads, 0, stream>>>(x, agg_out, agg_back, n4);
    }
    // 2) bidirectional scatter-add
    {
        long long total = E * 16;          // (edge, 4-channel group)
        int threads = 256;
        long long blocks = (total + threads - 1) / threads;
        biconv_scatter<<<(unsigned)blocks, threads, 0, stream>>>(
            x, sources, targets, agg_out, agg_back, total);
    }
    // 3) fused scale + dual GEMM + add via fp32 WMMA
    {
        int numRowTiles = (N + 15) >> 4;
        int numTiles = numRowTiles * 4;    // 4 column tiles of 16 (C=64)
        int wavesPerBlock = 8;             // 256 threads
        int blocks = (numTiles + wavesPerBlock - 1) / wavesPerBlock;
        biconv_wmma_gemm<<<blocks, 256, 0, stream>>>(
            agg_out, agg_back, norm, norm_t, w_out, w_back, out, N, numTiles);
    }
}